// LanczosNet_38809324486709
// MI455X (gfx1250) — compile-verified
//
#include <hip/hip_runtime.h>
#include <hip/hip_bf16.h>

typedef __attribute__((ext_vector_type(2))) float v2f;
typedef __attribute__((ext_vector_type(8))) float v8f;

#define WMMA_F32_4(a, b, c) \
    __builtin_amdgcn_wmma_f32_16x16x4_f32(false, (a), false, (b), (short)0, (c), false, false)

// ---------------------------------------------------------------------------
// t[30*60] = flatten(concat([Tri, Tri@Tri], axis=1))
// ---------------------------------------------------------------------------
__global__ void prep_t(const float* __restrict__ Tri, float* __restrict__ t) {
    for (int idx = threadIdx.x; idx < 30 * 60; idx += blockDim.x) {
        int i = idx / 60, j = idx % 60;
        float v;
        if (j < 30) {
            v = Tri[i * 30 + j];
        } else {
            int jj = j - 30;
            float s = 0.f;
            for (int m = 0; m < 30; ++m) s = fmaf(Tri[i * 30 + m], Tri[m * 30 + jj], s);
            v = s;
        }
        t[idx] = v;
    }
}

// ---------------------------------------------------------------------------
// GEMV: y[o] = act( sum_k x[k]*W[k,o] + b[o] ), W row-major [K, Nout].
// Chunked so the prefetch is once per 64 iters, inner loop branch-free.
// ---------------------------------------------------------------------------
__global__ void gemv_act(const float* __restrict__ x, const float* __restrict__ W,
                         const float* __restrict__ b, float* __restrict__ y,
                         int K, int Nout, int do_relu) {
    int o = blockIdx.x * blockDim.x + threadIdx.x;
    if (o >= Nout) return;
    float acc = b[o];
    int k0 = 0;
    for (; k0 + 64 <= K; k0 += 64) {
        if (k0 + 64 < K)
            __builtin_prefetch(&W[(size_t)(k0 + 64) * Nout + o], 0, 0);  // global_prefetch_b8
        #pragma unroll 8
        for (int k = k0; k < k0 + 64; ++k)
            acc = fmaf(x[k], W[(size_t)k * Nout + o], acc);
    }
    for (int k = k0; k < K; ++k)
        acc = fmaf(x[k], W[(size_t)k * Nout + o], acc);
    if (do_relu) acc = acc > 0.f ? acc : 0.f;
    y[o] = acc;
}

// ---------------------------------------------------------------------------
// Pack Q [4096,30] -> Qt [32,4096] transposed, rows 30/31 zeroed.
// One thread per source row k; writes coalesced across threads.
// ---------------------------------------------------------------------------
__global__ void pack_qt(const float* __restrict__ Qr, const float* __restrict__ Qi,
                        float* __restrict__ QrT, float* __restrict__ QiT) {
    int k = blockIdx.x * blockDim.x + threadIdx.x;
    if (k >= 4096) return;
    #pragma unroll
    for (int m = 0; m < 32; ++m) {
        QrT[(size_t)m * 4096 + k] = (m < 30) ? Qr[(size_t)k * 30 + m] : 0.0f;
        QiT[(size_t)m * 4096 + k] = (m < 30) ? Qi[(size_t)k * 30 + m] : 0.0f;
    }
}

// ---------------------------------------------------------------------------
// DDp_s[k,l] (32x32, zero-padded) = 0.5*(DDraw[k,l,s] + DDraw[l,k,s])
// ---------------------------------------------------------------------------
__global__ void symmetrize_pad(const float* __restrict__ DDraw,
                               float* __restrict__ DD0p, float* __restrict__ DD1p) {
    int idx = blockIdx.x * blockDim.x + threadIdx.x;
    if (idx >= 1024) return;
    int k = idx >> 5, l = idx & 31;
    float v0 = 0.f, v1 = 0.f;
    if (k < 30 && l < 30) {
        v0 = 0.5f * (DDraw[(k * 30 + l) * 2 + 0] + DDraw[(l * 30 + k) * 2 + 0]);
        v1 = 0.5f * (DDraw[(k * 30 + l) * 2 + 1] + DDraw[(l * 30 + k) * 2 + 1]);
    }
    DD0p[idx] = v0;
    DD1p[idx] = v1;
}

// ---------------------------------------------------------------------------
// U = Rr + Ii ; V = Ir - Ri   (each [32,64] flat; pad rows already zero)
// ---------------------------------------------------------------------------
__global__ void make_uv(const float* __restrict__ Rr, const float* __restrict__ Ri,
                        const float* __restrict__ Ir, const float* __restrict__ Ii,
                        float* __restrict__ U, float* __restrict__ V) {
    int i = blockIdx.x * blockDim.x + threadIdx.x;
    if (i >= 32 * 64) return;
    U[i] = Rr[i] + Ii[i];
    V[i] = Ir[i] - Ri[i];
}

// ---------------------------------------------------------------------------
// fp32 GEMM, V_WMMA_F32_16X16X4_F32. C[M,N] = A[M,K] @ B[K,N] (+C if acc).
// REQUIRES: M,N multiples of 16, K multiple of 4, lda even (v2f A loads).
// One wave = one 16x16 tile; branch-free K-loop, unroll 4 (16 WMMA-K/iter).
// grid = (N/16, M/16), block = 32.
// ---------------------------------------------------------------------------
__global__ void gemm_wmma_pad(const float* __restrict__ A, int lda,
                              const float* __restrict__ B, int ldb,
                              float* __restrict__ C, int ldc,
                              int K, int acc) {
    const int lane = threadIdx.x & 31;
    const int half = lane >> 4;     // fp32 WMMA A/B K sub-group
    const int li   = lane & 15;
    const int m0 = blockIdx.y * 16;
    const int n0 = blockIdx.x * 16;
    const float* Ap = A + (size_t)(m0 + li) * lda;  // lane's A row
    const float* Bp = B + (n0 + li);                // lane's B column

    v8f c = {};
    if (acc) {
        #pragma unroll
        for (int r = 0; r < 8; ++r)
            c[r] = C[(size_t)(m0 + r + 8 * half) * ldc + (n0 + li)];
    }
    #pragma unroll 4
    for (int k0 = 0; k0 < K; k0 += 4) {
        int ka = k0 + 2 * half;                     // even -> 8B-aligned A load
        v2f a = *(const v2f*)(Ap + ka);
        v2f b;
        b.x = Bp[(size_t)ka * ldb];
        b.y = Bp[(size_t)(ka + 1) * ldb];
        c = WMMA_F32_4(a, b, c);
    }
    #pragma unroll
    for (int r = 0; r < 8; ++r)
        C[(size_t)(m0 + r + 8 * half) * ldc + (n0 + li)] = c[r];
}

// ---------------------------------------------------------------------------
// Fused output: X[:, :64] = Qr@P + Qi@Pq + b ; X[:, 64:] = Qi@P - Qr@Pq + b
// A from QrT/QiT [32,4096] (padded transpose => A(m,k)=QT[k,row]);
// P,Pq [32,64] with zero pad rows. K=32 branch-free, 4 accumulators.
// grid = (4, 256), block = 32.
// ---------------------------------------------------------------------------
__global__ void fuse_out(const float* __restrict__ QrT, const float* __restrict__ QiT,
                         const float* __restrict__ P, const float* __restrict__ Pq,
                         const float* __restrict__ conv_b, float* __restrict__ X) {
    const int lane = threadIdx.x & 31;
    const int half = lane >> 4;
    const int li   = lane & 15;
    const int m0 = blockIdx.y * 16;
    const int n0 = blockIdx.x * 16;
    const int row = m0 + li;   // < 4096
    const int col = n0 + li;   // < 64

    v8f cR1 = {}, cR2 = {}, cI1 = {}, cI2 = {};
    #pragma unroll
    for (int k0 = 0; k0 < 32; k0 += 4) {
        int ka = k0 + 2 * half;
        v2f aR, aI, bP, bQ;
        aR.x = QrT[(size_t)ka * 4096 + row];
        aR.y = QrT[(size_t)(ka + 1) * 4096 + row];
        aI.x = QiT[(size_t)ka * 4096 + row];
        aI.y = QiT[(size_t)(ka + 1) * 4096 + row];
        bP.x = P [(size_t)ka * 64 + col];
        bP.y = P [(size_t)(ka + 1) * 64 + col];
        bQ.x = Pq[(size_t)ka * 64 + col];
        bQ.y = Pq[(size_t)(ka + 1) * 64 + col];
        cR1 = WMMA_F32_4(aR, bP, cR1);
        cR2 = WMMA_F32_4(aI, bQ, cR2);
        cI1 = WMMA_F32_4(aI, bP, cI1);
        cI2 = WMMA_F32_4(aR, bQ, cI2);
    }
    float bias = conv_b[col];
    #pragma unroll
    for (int r = 0; r < 8; ++r) {
        int mm = m0 + r + 8 * half;
        X[(size_t)mm * 128 + col]      = cR1[r] + cR2[r] + bias;  // out_real
        X[(size_t)mm * 128 + 64 + col] = cI1[r] - cI2[r] + bias;  // out_imag
    }
}

// ---------------------------------------------------------------------------
// Head: y[o,n] = fw[o,:]·X[n,:] + fb[o]; log_softmax over o (7); out[o*4096+n]
// ---------------------------------------------------------------------------
__global__ void head_logsoftmax(const float* __restrict__ X, const float* __restrict__ fw,
                                const float* __restrict__ fb, float* __restrict__ out) {
    int n = blockIdx.x * blockDim.x + threadIdx.x;
    if (n >= 4096) return;
    float y[7];
    float mx = -3.0e38f;
    #pragma unroll
    for (int o = 0; o < 7; ++o) {
        float acc = fb[o];
        for (int c = 0; c < 128; ++c)
            acc = fmaf(fw[o * 128 + c], X[(size_t)n * 128 + c], acc);
        y[o] = acc;
        mx = fmaxf(mx, acc);
    }
    float sum = 0.f;
    #pragma unroll
    for (int o = 0; o < 7; ++o) sum += __expf(y[o] - mx);
    float lse = mx + __logf(sum);
    #pragma unroll
    for (int o = 0; o < 7; ++o) out[(size_t)o * 4096 + n] = y[o] - lse;
}

// ---------------------------------------------------------------------------
// Launcher
// ---------------------------------------------------------------------------
extern "C" void kernel_launch(void* const* d_in, const int* in_sizes, int n_in,
                              void* d_out, int out_size, void* d_ws, size_t ws_size,
                              hipStream_t stream) {
    const float* real   = (const float*)d_in[0];   // [4096,64]
    const float* imag   = (const float*)d_in[1];   // [4096,64]
    const float* Tri    = (const float*)d_in[2];   // [30,30]
    const float* Qr     = (const float*)d_in[3];   // [4096,30]
    const float* Qi     = (const float*)d_in[4];   // [4096,30]
    const float* w1     = (const float*)d_in[5];   // [1800,1024]
    const float* b1     = (const float*)d_in[6];
    const float* w2     = (const float*)d_in[7];   // [1024,1024]
    const float* b2     = (const float*)d_in[8];
    const float* w3     = (const float*)d_in[9];   // [1024,1024]
    const float* b3     = (const float*)d_in[10];
    const float* w4     = (const float*)d_in[11];  // [1024,1800]
    const float* b4     = (const float*)d_in[12];
    const float* conv_w = (const float*)d_in[13];  // [2,64,64]
    const float* conv_b = (const float*)d_in[14];  // [1,64]
    const float* flat_w = (const float*)d_in[15];  // [7,128,1]
    const float* flat_b = (const float*)d_in[16];  // [7]
    float* out = (float*)d_out;                    // [7,4096]
    float* ws  = (float*)d_ws;

    // workspace layout (floats; all offsets even for v2f alignment)
    float* t     = ws;              // 1800
    float* h1    = ws + 1800;       // 1024
    float* h2    = ws + 2824;       // 1024
    float* h3    = ws + 3848;       // 1024
    float* DDraw = ws + 4872;       // 1800
    float* DD0p  = ws + 6672;       // 1024  [32,32] padded
    float* DD1p  = ws + 7696;       // 1024
    float* QrT   = ws + 8720;       // 131072 [32,4096] padded transpose
    float* QiT   = ws + 139792;     // 131072
    float* Rr    = ws + 270864;     // 2048  [32,64]
    float* Ri    = ws + 272912;     // 2048
    float* Ir    = ws + 274960;     // 2048
    float* Ii    = ws + 277008;     // 2048
    float* U     = ws + 279056;     // 2048
    float* V     = ws + 281104;     // 2048
    float* T0    = ws + 283152;     // 2048
    float* T1    = ws + 285200;     // 2048
    float* S0    = ws + 287248;     // 2048
    float* S1    = ws + 289296;     // 2048
    float* P     = ws + 291344;     // 2048
    float* Pq    = ws + 293392;     // 2048
    float* X     = ws + 295440;     // 524288 [4096,128]

    dim3 b32(32), b256(256);
    dim3 gSmall(4, 2);   // [32 x 64] tiles

    // --- data repacking (independent of MLP; issue first) ---
    pack_qt<<<16, b256, 0, stream>>>(Qr, Qi, QrT, QiT);

    // --- spectral filter MLP (tiny GEMV chain) ---
    prep_t<<<1, b256, 0, stream>>>(Tri, t);
    gemv_act<<<4, b256, 0, stream>>>(t,  w1, b1, h1, 1800, 1024, 1);
    gemv_act<<<4, b256, 0, stream>>>(h1, w2, b2, h2, 1024, 1024, 1);
    gemv_act<<<4, b256, 0, stream>>>(h2, w3, b3, h3, 1024, 1024, 1);
    gemv_act<<<8, b256, 0, stream>>>(h3, w4, b4, DDraw, 1024, 1800, 0);
    symmetrize_pad<<<4, b256, 0, stream>>>(DDraw, DD0p, DD1p);

    // --- projections: [32,64] = QT[32,4096] @ X[4096,64]  (K = 4096) ---
    gemm_wmma_pad<<<gSmall, b32, 0, stream>>>(QrT, 4096, real, 64, Rr, 64, 4096, 0);
    gemm_wmma_pad<<<gSmall, b32, 0, stream>>>(QrT, 4096, imag, 64, Ri, 64, 4096, 0);
    gemm_wmma_pad<<<gSmall, b32, 0, stream>>>(QiT, 4096, real, 64, Ir, 64, 4096, 0);
    gemm_wmma_pad<<<gSmall, b32, 0, stream>>>(QiT, 4096, imag, 64, Ii, 64, 4096, 0);
    make_uv<<<8, b256, 0, stream>>>(Rr, Ri, Ir, Ii, U, V);

    // --- small chains: T_s = DDp_s @ U, S_s = DDp_s @ V; P/Pq = Σ_s (·) @ W_s ---
    gemm_wmma_pad<<<gSmall, b32, 0, stream>>>(DD0p, 32, U, 64, T0, 64, 32, 0);
    gemm_wmma_pad<<<gSmall, b32, 0, stream>>>(DD1p, 32, U, 64, T1, 64, 32, 0);
    gemm_wmma_pad<<<gSmall, b32, 0, stream>>>(DD0p, 32, V, 64, S0, 64, 32, 0);
    gemm_wmma_pad<<<gSmall, b32, 0, stream>>>(DD1p, 32, V, 64, S1, 64, 32, 0);
    gemm_wmma_pad<<<gSmall, b32, 0, stream>>>(T0, 64, conv_w,        64, P,  64, 64, 0);
    gemm_wmma_pad<<<gSmall, b32, 0, stream>>>(T1, 64, conv_w + 4096, 64, P,  64, 64, 1);
    gemm_wmma_pad<<<gSmall, b32, 0, stream>>>(S0, 64, conv_w,        64, Pq, 64, 64, 0);
    gemm_wmma_pad<<<gSmall, b32, 0, stream>>>(S1, 64, conv_w + 4096, 64, Pq, 64, 64, 1);

    // --- fused output GEMM + bias + concat (4 WMMA accumulators / tile) ---
    fuse_out<<<dim3(4, 256), b32, 0, stream>>>(QrT, QiT, P, Pq, conv_b, X);

    // --- head + log_softmax ---
    head_logsoftmax<<<16, b256, 0, stream>>>(X, flat_w, flat_b, out);
}